// SaliencyMapBasedGlimpsing_85813446574375
// MI455X (gfx1250) — compile-verified
//
#include <hip/hip_runtime.h>
#include <stdint.h>

#define NPIX 1024
#define NN (NPIX * NPIX)
#define NB 16
#define TPB 256
#define PIX_PER_BLOCK 4096
#define BLOCKS_PER_BATCH (NN / PIX_PER_BLOCK)   // 256
#define TOTAL_BLOCKS (NB * BLOCKS_PER_BATCH)    // 4096
#define ITERS 4                                  // 1024 px per iter
#define RADIUS 4.76837158203125e-07f             // 1 / 1024^2 / 2
#define F_INF __builtin_inff()

struct Partial {
  float maxA, dA; unsigned iA;   // variant A: masked = hist * sal
  float maxB, dB; unsigned iB;   // variant B: masked = sal (hist replaced by ones)
  unsigned flags;                // bit0: anyA!=0, bit1: anyB!=0, bit2: any hist!=0
  unsigned pad;
};

// Low 32 bits of a generic pointer to __shared__ == byte offset into the wave's LDS.
__device__ __forceinline__ uint32_t lds_off(const void* p) {
  return (uint32_t)(uintptr_t)p;
}

// CDNA5 async copy: global -> LDS, 16 bytes per lane, tracked with ASYNCcnt.
__device__ __forceinline__ void async_copy_b128(uint32_t lds, const void* g) {
  asm volatile("global_load_async_to_lds_b128 %0, %1, off"
               :: "v"(lds), "v"(g) : "memory");
}

// ---------------------------------------------------------------------------
// Pass 1: fused IOR-mask + max + candidate-argmin reduction, dual variant.
// Double-buffered async global->LDS staging (sal, hist, grid tiles).
// ---------------------------------------------------------------------------
__global__ void __launch_bounds__(TPB)
k_reduce(const float* __restrict__ sal, const float* __restrict__ hist,
         const float* __restrict__ cur, const float* __restrict__ grid,
         Partial* __restrict__ parts)
{
  __shared__ float sb[2][1024];
  __shared__ float hb[2][1024];
  __shared__ float gb[2][2048];
  __shared__ float rmA[TPB], rdA[TPB], rmB[TPB], rdB[TPB];
  __shared__ unsigned riA[TPB], riB[TPB], rfl[TPB];

  const int tid   = threadIdx.x;
  const int blk   = blockIdx.x;
  const int batch = blk / BLOCKS_PER_BATCH;
  const int chunk = blk - batch * BLOCKS_PER_BATCH;
  const int imgBase = chunk * PIX_PER_BLOCK;          // pixel index inside image
  const float* salB  = sal  + (size_t)batch * NN;
  const float* histB = hist + (size_t)batch * NN;
  const float cx = cur[batch * 2 + 0];
  const float cy = cur[batch * 2 + 1];

  // stage iteration 0
  {
    const int p = imgBase + tid * 4;
    async_copy_b128(lds_off(&sb[0][tid * 4]), salB + p);
    async_copy_b128(lds_off(&hb[0][tid * 4]), histB + p);
    async_copy_b128(lds_off(&gb[0][tid * 8]),     grid + (size_t)p * 2);
    async_copy_b128(lds_off(&gb[0][tid * 8 + 4]), grid + (size_t)p * 2 + 4);
  }

  float mA = -F_INF, dA = F_INF; unsigned iA = 0u;
  float mB = -F_INF, dB = F_INF; unsigned iB = 0u;
  unsigned fl = 0u;

  #pragma unroll
  for (int it = 0; it < ITERS; ++it) {
    if (it + 1 < ITERS) {
      const int buf = (it + 1) & 1;
      const int p = imgBase + (it + 1) * 1024 + tid * 4;
      async_copy_b128(lds_off(&sb[buf][tid * 4]), salB + p);
      async_copy_b128(lds_off(&hb[buf][tid * 4]), histB + p);
      async_copy_b128(lds_off(&gb[buf][tid * 8]),     grid + (size_t)p * 2);
      async_copy_b128(lds_off(&gb[buf][tid * 8 + 4]), grid + (size_t)p * 2 + 4);
      // async loads complete in order: <=4 outstanding => current buffer landed
      asm volatile("s_wait_asynccnt 0x4" ::: "memory");
    } else {
      asm volatile("s_wait_asynccnt 0x0" ::: "memory");
    }
    const int buf = it & 1;
    const float4 sv  = *(const float4*)&sb[buf][tid * 4];
    const float4 hv  = *(const float4*)&hb[buf][tid * 4];
    const float4 g01 = *(const float4*)&gb[buf][tid * 8];
    const float4 g23 = *(const float4*)&gb[buf][tid * 8 + 4];
    const unsigned pbase = (unsigned)(imgBase + it * 1024 + tid * 4);

    const float ss[4] = {sv.x, sv.y, sv.z, sv.w};
    const float hh[4] = {hv.x, hv.y, hv.z, hv.w};
    const float gx[4] = {g01.x, g01.z, g23.x, g23.z};
    const float gy[4] = {g01.y, g01.w, g23.y, g23.w};

    #pragma unroll
    for (int j = 0; j < 4; ++j) {
      const float ddx = gx[j] - cx, ddy = gy[j] - cy;
      const float cd = 0.5f * (ddx * ddx + ddy * ddy);   // == cur_dist == argmin d
      const float sm = (cd <= RADIUS) ? 0.0f : ss[j];    // sal after IOR circle
      const float h  = hh[j];
      const float vA = h * sm;
      const float vB = sm;
      const unsigned idx = pbase + j;
      fl |= (vA != 0.0f) ? 1u : 0u;
      fl |= (vB != 0.0f) ? 2u : 0u;
      fl |= (h  != 0.0f) ? 4u : 0u;
      // lexicographic (max value, min d, min idx) trackers
      if (vA > mA) { mA = vA; const bool q = vA > 0.0f; dA = q ? cd : F_INF; iA = q ? idx : 0u; }
      else if (vA == mA && vA > 0.0f) {
        if (cd < dA || (cd == dA && idx < iA)) { dA = cd; iA = idx; }
      }
      if (vB > mB) { mB = vB; const bool q = vB > 0.0f; dB = q ? cd : F_INF; iB = q ? idx : 0u; }
      else if (vB == mB && vB > 0.0f) {
        if (cd < dB || (cd == dB && idx < iB)) { dB = cd; iB = idx; }
      }
    }
  }

  rmA[tid] = mA; rdA[tid] = dA; riA[tid] = iA;
  rmB[tid] = mB; rdB[tid] = dB; riB[tid] = iB;
  rfl[tid] = fl;
  __syncthreads();
  for (int s = TPB / 2; s > 0; s >>= 1) {
    if (tid < s) {
      const int o = tid + s;
      { const float m2 = rmA[o];
        if (m2 > rmA[tid]) { rmA[tid] = m2; rdA[tid] = rdA[o]; riA[tid] = riA[o]; }
        else if (m2 == rmA[tid]) {
          const float d2 = rdA[o]; const unsigned i2 = riA[o];
          if (d2 < rdA[tid] || (d2 == rdA[tid] && i2 < riA[tid])) { rdA[tid] = d2; riA[tid] = i2; }
        } }
      { const float m2 = rmB[o];
        if (m2 > rmB[tid]) { rmB[tid] = m2; rdB[tid] = rdB[o]; riB[tid] = riB[o]; }
        else if (m2 == rmB[tid]) {
          const float d2 = rdB[o]; const unsigned i2 = riB[o];
          if (d2 < rdB[tid] || (d2 == rdB[tid] && i2 < riB[tid])) { rdB[tid] = d2; riB[tid] = i2; }
        } }
      rfl[tid] |= rfl[o];
    }
    __syncthreads();
  }
  if (tid == 0) {
    Partial P;
    P.maxA = rmA[0]; P.dA = rdA[0]; P.iA = riA[0];
    P.maxB = rmB[0]; P.dB = rdB[0]; P.iB = riB[0];
    P.flags = rfl[0]; P.pad = 0u;
    parts[blk] = P;
  }
}

// ---------------------------------------------------------------------------
// Pass 2a: global "any hist != 0" flag (jnp.all(hist==0) is over the whole tensor)
// ---------------------------------------------------------------------------
__global__ void __launch_bounds__(TPB)
k_flag(const Partial* __restrict__ parts, unsigned* __restrict__ gflag)
{
  __shared__ unsigned sf[TPB];
  unsigned f = 0u;
  for (int i = threadIdx.x; i < TOTAL_BLOCKS; i += TPB) f |= parts[i].flags;
  sf[threadIdx.x] = f;
  __syncthreads();
  for (int s = TPB / 2; s > 0; s >>= 1) {
    if (threadIdx.x < s) sf[threadIdx.x] |= sf[threadIdx.x + s];
    __syncthreads();
  }
  if (threadIdx.x == 0) *gflag = sf[0];
}

// ---------------------------------------------------------------------------
// Pass 2b: per-batch combine -> winner idx, next_xy_location, {loc, eps, replace}
// ---------------------------------------------------------------------------
__global__ void __launch_bounds__(TPB)
k_combine(const Partial* __restrict__ parts, const unsigned* __restrict__ gflag,
          const float* __restrict__ sal, const float* __restrict__ grid,
          const float* __restrict__ cur, float* __restrict__ outLoc,
          float4* __restrict__ params)
{
  __shared__ float rm[TPB], rd[TPB];
  __shared__ unsigned ri[TPB], rany[TPB];
  const int tid = threadIdx.x;
  const int b = blockIdx.x;
  const unsigned histNZ = (*gflag) & 4u;
  const Partial P = parts[b * BLOCKS_PER_BATCH + tid];
  if (histNZ) { rm[tid] = P.maxA; rd[tid] = P.dA; ri[tid] = P.iA; rany[tid] = P.flags & 1u; }
  else        { rm[tid] = P.maxB; rd[tid] = P.dB; ri[tid] = P.iB; rany[tid] = (P.flags >> 1) & 1u; }
  __syncthreads();
  for (int s = TPB / 2; s > 0; s >>= 1) {
    if (tid < s) {
      const int o = tid + s;
      const float m2 = rm[o];
      if (m2 > rm[tid]) { rm[tid] = m2; rd[tid] = rd[o]; ri[tid] = ri[o]; }
      else if (m2 == rm[tid]) {
        const float d2 = rd[o]; const unsigned i2 = ri[o];
        if (d2 < rd[tid] || (d2 == rd[tid] && i2 < ri[tid])) { rd[tid] = d2; ri[tid] = i2; }
      }
      rany[tid] |= rany[o];
    }
    __syncthreads();
  }
  if (tid == 0) {
    const unsigned bi = ri[0];               // no candidate anywhere -> 0 (matches argmin of all-inf)
    const unsigned px = bi & (NPIX - 1);
    const unsigned py = bi >> 10;
    const float lx = 2.0f * ((float)(px + 1) * (1.0f / NPIX)) - 1.0f;
    const float ly = 2.0f * ((float)(py + 1) * (1.0f / NPIX)) - 1.0f;
    const float cx = cur[b * 2 + 0], cy = cur[b * 2 + 1];
    const float sv = sal[(size_t)b * NN + bi];
    const float gx = grid[(size_t)bi * 2 + 0], gy = grid[(size_t)bi * 2 + 1];
    const float ddx = gx - cx, ddy = gy - cy;
    const float cd = 0.5f * (ddx * ddx + ddy * ddy);
    float sm = (cd <= RADIUS) ? 0.0f : sv;   // sm_value from post-IOR sal
    sm = fmaxf(sm, 0.0001f);
    const float eps = 10.0f / sm;
    const bool allz = (rany[0] == 0u);
    outLoc[b * 2 + 0] = allz ? 1.0f : lx;
    outLoc[b * 2 + 1] = allz ? 1.0f : ly;
    // rbf/hist update always use un-overridden loc (reference semantics)
    params[b] = make_float4(lx, ly, eps, histNZ ? 0.0f : 1.0f);
  }
}

// ---------------------------------------------------------------------------
// Pass 3: elementwise RBF mask + hist update, float4-vectorized streaming.
// ---------------------------------------------------------------------------
__global__ void __launch_bounds__(TPB)
k_apply(const float* __restrict__ hist, const float* __restrict__ grid,
        const float4* __restrict__ params,
        float* __restrict__ outHist, float* __restrict__ outMask)
{
  const long long t = (long long)blockIdx.x * TPB + threadIdx.x;
  const long long e = t * 4;                 // flat element over B*NN
  const int b   = (int)(e >> 20);            // e / NN
  const int pix = (int)(e & (NN - 1));
  const float4 prm = params[b];
  const float lx = prm.x, ly = prm.y, eps = prm.z, replace = prm.w;
  const float4 hv  = *(const float4*)(hist + e);
  const float4 g01 = *(const float4*)(grid + (size_t)pix * 2);
  const float4 g23 = *(const float4*)(grid + (size_t)pix * 2 + 4);
  const float hh[4] = {hv.x, hv.y, hv.z, hv.w};
  const float gx[4] = {g01.x, g01.z, g23.x, g23.z};
  const float gy[4] = {g01.y, g01.w, g23.y, g23.w};
  float m_[4], h_[4];
  #pragma unroll
  for (int j = 0; j < 4; ++j) {
    const float ddx = gx[j] - lx, ddy = gy[j] - ly;
    const float rbf = 0.5f * (ddx * ddx + ddy * ddy);
    const float a = rbf * eps;
    const float mk = __expf(-(a * a));
    const float he = (replace != 0.0f) ? 1.0f : hh[j];
    float hn = he * (1.0f - mk);
    hn = fminf(fmaxf(hn, 0.0f), 1.0f);
    m_[j] = mk; h_[j] = hn;
  }
  *(float4*)(outHist + e) = make_float4(h_[0], h_[1], h_[2], h_[3]);
  *(float4*)(outMask + e) = make_float4(m_[0], m_[1], m_[2], m_[3]);
}

// ---------------------------------------------------------------------------
extern "C" void kernel_launch(void* const* d_in, const int* in_sizes, int n_in,
                              void* d_out, int out_size, void* d_ws, size_t ws_size,
                              hipStream_t stream)
{
  (void)in_sizes; (void)n_in; (void)out_size; (void)ws_size;
  const float* sal  = (const float*)d_in[0];   // (16,1024,1024)
  const float* hist = (const float*)d_in[1];   // (16,1,1024,1024)
  const float* cur  = (const float*)d_in[2];   // (16,2)
  const float* grid = (const float*)d_in[3];   // (1024,1024,2)

  float* out     = (float*)d_out;
  float* outLoc  = out;                                   // 32 floats
  float* outHist = out + 2 * NB;                          // 16M floats
  float* outMask = out + 2 * NB + (size_t)NB * NN;        // 16M floats

  char* ws = (char*)d_ws;
  Partial*  parts  = (Partial*)ws;                                        // 4096*32 B
  unsigned* gflag  = (unsigned*)(ws + sizeof(Partial) * TOTAL_BLOCKS);    // 4 B
  float4*   params = (float4*)(ws + sizeof(Partial) * TOTAL_BLOCKS + 16); // 16*16 B

  hipLaunchKernelGGL(k_reduce, dim3(TOTAL_BLOCKS), dim3(TPB), 0, stream,
                     sal, hist, cur, grid, parts);
  hipLaunchKernelGGL(k_flag, dim3(1), dim3(TPB), 0, stream, parts, gflag);
  hipLaunchKernelGGL(k_combine, dim3(NB), dim3(TPB), 0, stream,
                     parts, gflag, sal, grid, cur, outLoc, params);
  hipLaunchKernelGGL(k_apply, dim3((NB * (NN / 4)) / TPB), dim3(TPB), 0, stream,
                     hist, grid, params, outHist, outMask);
}